// GCNN_26628797236068
// MI455X (gfx1250) — compile-verified
//
#include <hip/hip_runtime.h>
#include <math.h>

typedef __attribute__((ext_vector_type(2))) float v2f;
typedef __attribute__((ext_vector_type(8))) float v8f;

#define CHANNELS 16
#define KDIM 128

// ---------------------------------------------------------------------------
// Kernel 1: init — zero the output accumulator (d_out is reused as the
// scatter-add buffer) and set deg[i] = 1.0 (accounts for the self-loop).
// ---------------------------------------------------------------------------
__global__ void gcn_init_kernel(float* __restrict__ out_accum,
                                float* __restrict__ deg, int n) {
    int i = blockIdx.x * blockDim.x + threadIdx.x;
    if (i < n * CHANNELS) out_accum[i] = 0.0f;
    if (i < n) deg[i] = 1.0f;
}

// ---------------------------------------------------------------------------
// Kernel 2: h = x @ W with V_WMMA_F32_16X16X4_F32.
// One wave per 16-row tile of h. K = 128 -> 32 WMMA steps.
//
// Per ISA layouts (CDNA5 ISA):
//   A (16x4 f32, MxK): lanes 0-15 m=lane, v0=K0,v1=K1; lanes 16-31 m=lane-16,
//   v0=K2,v1=K3  => per-lane contiguous float2 load at x[m, k + 2*(lane>>4)].
//   B (4x16 f32, KxN): lanes 0-15 n=lane hold K0/K1; lanes 16-31 hold K2/K3.
//   C/D (16x16 f32): vgpr v, lanes 0-15 -> row v, col=lane;
//                    lanes 16-31 -> row v+8, col=lane-16.
// ---------------------------------------------------------------------------
__global__ void gcn_gemm_wmma_kernel(const float* __restrict__ x,
                                     const float* __restrict__ W,
                                     float* __restrict__ h, int n) {
    const int lane   = threadIdx.x & 31;
    const int waveId = threadIdx.x >> 5;
    const int ntiles = (n + 15) >> 4;
    const int tile   = blockIdx.x * (blockDim.x >> 5) + waveId;
    if (tile >= ntiles) return;   // uniform per-wave: EXEC stays all-ones

    const int half = lane >> 4;        // 0: lanes 0-15, 1: lanes 16-31
    const int idx  = lane & 15;        // m for A, n for B/C
    const int koff = half * 2;

    int arow = tile * 16 + idx;
    if (arow >= n) arow = n - 1;       // clamp (N=100000 is 16-aligned anyway)
    const float* xrow = x + (size_t)arow * KDIM;

    v8f c = {};
#pragma unroll
    for (int k = 0; k < KDIM; k += 4) {
        // A fragment: contiguous float2 from x
        v2f a;
        a.x = xrow[k + koff + 0];
        a.y = xrow[k + koff + 1];
        // B fragment: W is [128 x 16] row-major
        v2f bm;
        bm.x = W[(k + koff + 0) * CHANNELS + idx];
        bm.y = W[(k + koff + 1) * CHANNELS + idx];
        c = __builtin_amdgcn_wmma_f32_16x16x4_f32(
            /*neg_a=*/false, a, /*neg_b=*/false, bm,
            /*c_mod=*/(short)0, c, /*reuse_a=*/false, /*reuse_b=*/false);
    }

    // Store D: row = v + 8*half, col = idx
#pragma unroll
    for (int v = 0; v < 8; ++v) {
        int row = tile * 16 + v + 8 * half;
        if (row < n) h[(size_t)row * CHANNELS + idx] = c[v];
    }
}

// ---------------------------------------------------------------------------
// Kernel 3: degree over targets (deg already holds 1.0 for the self-loop).
// ---------------------------------------------------------------------------
__global__ void gcn_degree_kernel(const int* __restrict__ col,
                                  float* __restrict__ deg, int e) {
    int i = blockIdx.x * blockDim.x + threadIdx.x;
    if (i < e) atomicAdd(&deg[col[i]], 1.0f);
}

// ---------------------------------------------------------------------------
// Kernel 4: dis[i] = rsqrt(deg[i])  (deg >= 1 always, matches where(deg>0)).
// ---------------------------------------------------------------------------
__global__ void gcn_rsqrt_kernel(float* __restrict__ deg, int n) {
    int i = blockIdx.x * blockDim.x + threadIdx.x;
    if (i < n) deg[i] = rsqrtf(deg[i]);
}

// ---------------------------------------------------------------------------
// Kernel 5: edge scatter. One thread per (edge, channel); 16 consecutive
// lanes share an edge so the h-gather and the out-atomics are contiguous
// 64B bursts that live in the 192MB L2 (h and out are 6.4MB each).
// ---------------------------------------------------------------------------
__global__ void gcn_scatter_kernel(const int* __restrict__ row,
                                   const int* __restrict__ col,
                                   const float* __restrict__ dis,
                                   const float* __restrict__ h,
                                   float* __restrict__ out_accum, int e) {
    long long t = (long long)blockIdx.x * blockDim.x + threadIdx.x;
    long long ec = (long long)e * CHANNELS;
    if (t >= ec) return;
    int edge = (int)(t >> 4);
    int ch   = (int)(t & 15);
    int r = row[edge];
    int cl = col[edge];
    float norm = dis[r] * dis[cl];
    float msg  = norm * h[(size_t)r * CHANNELS + ch];
    atomicAdd(&out_accum[(size_t)cl * CHANNELS + ch], msg);
}

// ---------------------------------------------------------------------------
// Kernel 6: per-node epilogue: add self-loop term + bias, relu, log_softmax.
// One thread per node over 16 channels (registers only).
// ---------------------------------------------------------------------------
__global__ void gcn_final_kernel(float* __restrict__ out,
                                 const float* __restrict__ h,
                                 const float* __restrict__ dis,
                                 const float* __restrict__ b, int n) {
    int i = blockIdx.x * blockDim.x + threadIdx.x;
    if (i >= n) return;
    float selfw = dis[i] * dis[i];
    float v[CHANNELS];
    float m = -INFINITY;
#pragma unroll
    for (int c = 0; c < CHANNELS; ++c) {
        float val = out[(size_t)i * CHANNELS + c]
                  + selfw * h[(size_t)i * CHANNELS + c] + b[c];
        val = fmaxf(val, 0.0f);          // relu
        v[c] = val;
        m = fmaxf(m, val);
    }
    float s = 0.0f;
#pragma unroll
    for (int c = 0; c < CHANNELS; ++c) s += __expf(v[c] - m);
    float lse = m + __logf(s);
#pragma unroll
    for (int c = 0; c < CHANNELS; ++c)
        out[(size_t)i * CHANNELS + c] = v[c] - lse;
}

// ---------------------------------------------------------------------------
extern "C" void kernel_launch(void* const* d_in, const int* in_sizes, int n_in,
                              void* d_out, int out_size, void* d_ws, size_t ws_size,
                              hipStream_t stream) {
    const float* x  = (const float*)d_in[0];
    const int*   ei = (const int*)d_in[1];     // [2, E] flat: row then col
    const float* W  = (const float*)d_in[2];
    const float* b  = (const float*)d_in[3];
    float* out = (float*)d_out;

    const int n = in_sizes[0] / KDIM;          // 100000
    const int e = in_sizes[1] / 2;             // 3200000
    const int* row = ei;
    const int* col = ei + e;

    // workspace layout: h [n*16 f32] | deg/dis [n f32]
    float* h   = (float*)d_ws;
    float* deg = h + (size_t)n * CHANNELS;

    const int B = 256;

    // 1. init: zero out accumulator, deg = 1 (self-loop)
    gcn_init_kernel<<<(n * CHANNELS + B - 1) / B, B, 0, stream>>>(out, deg, n);

    // 2. h = x @ W  (WMMA f32 16x16x4, one wave per 16-row tile)
    {
        const int ntiles = (n + 15) / 16;
        const int wavesPerBlock = 4;           // 128 threads
        const int blocks = (ntiles + wavesPerBlock - 1) / wavesPerBlock;
        gcn_gemm_wmma_kernel<<<blocks, wavesPerBlock * 32, 0, stream>>>(x, W, h, n);
    }

    // 3. degree over targets
    gcn_degree_kernel<<<(e + B - 1) / B, B, 0, stream>>>(col, deg, e);

    // 4. deg -> deg^{-1/2}
    gcn_rsqrt_kernel<<<(n + B - 1) / B, B, 0, stream>>>(deg, n);

    // 5. normalized scatter-add over edges
    {
        long long work = (long long)e * CHANNELS;
        int blocks = (int)((work + B - 1) / B);
        gcn_scatter_kernel<<<blocks, B, 0, stream>>>(row, col, deg, h, out, e);
    }

    // 6. self-loop + bias + relu + log_softmax
    gcn_final_kernel<<<(n + B - 1) / B, B, 0, stream>>>(out, h, deg, b, n);
}